// GATSingleLayer_64845416235491
// MI455X (gfx1250) — compile-verified
//
#include <hip/hip_runtime.h>
#include <hip/hip_bf16.h>

#define N_NODES 50000
#define N_EDGES 800000
#define F_IN    128
#define HEADS   8
#define FH      16
#define HF      128           // HEADS*FH
#define NEG_SLOPE 0.2f

typedef float v2f __attribute__((ext_vector_type(2)));
typedef float v4f __attribute__((ext_vector_type(4)));
typedef float v8f __attribute__((ext_vector_type(8)));

// ---------------------------------------------------------------------------
// 1) h = x @ W^T via V_WMMA_F32_16X16X4_F32.
//    One wave per 16x16 output tile. Block = 256 threads = 8 waves; wave w
//    covers N-tile w (8*16 = 128 output features). grid.x = 50000/16 = 3125.
//    A-layout (16x4 f32): lanes 0-15 hold K=k0,k0+1; lanes 16-31 hold K=k0+2,k0+3.
//    B-layout (4x16 f32): VGPR v, lanes 0-15 -> K=v; lanes 16-31 -> K=v+2.
//    C/D: VGPR v, lanes 0-15 -> M=v, N=lane; lanes 16-31 -> M=v+8, N=lane-16.
// ---------------------------------------------------------------------------
__global__ __launch_bounds__(256) void gat_gemm_wmma(
    const float* __restrict__ x, const float* __restrict__ W,
    float* __restrict__ h) {
  const int lane = threadIdx.x & 31;
  const int wave = threadIdx.x >> 5;
  const int m0   = blockIdx.x * 16;
  const int n0   = wave * 16;
  const int half = lane >> 4;     // 0: K+0,K+1   1: K+2,K+3
  const int l15  = lane & 15;

  const float* __restrict__ arow = x + (m0 + l15) * F_IN + 2 * half;
  const float* __restrict__ brow = W + (n0 + l15) * F_IN + 2 * half; // B[k][j]=W[j][k]

  v8f c = {};
#pragma unroll
  for (int k0 = 0; k0 < F_IN; k0 += 4) {
    v2f a; a.x = arow[k0]; a.y = arow[k0 + 1];
    v2f b; b.x = brow[k0]; b.y = brow[k0 + 1];
    c = __builtin_amdgcn_wmma_f32_16x16x4_f32(
        /*neg_a=*/false, a, /*neg_b=*/false, b,
        /*c_mod=*/(short)0, c, /*reuse_a=*/false, /*reuse_b=*/false);
  }

  float* __restrict__ op = h + (m0 + half * 8) * HF + n0 + l15;
#pragma unroll
  for (int v = 0; v < 8; ++v) op[v * HF] = c[v];
}

// ---------------------------------------------------------------------------
// 2) Per-(node,head) attention logits: a_src/a_dst = <h[n,head,:], att_*[head,:]>
// ---------------------------------------------------------------------------
__global__ void gat_logits(const float* __restrict__ h,
                           const float* __restrict__ att_src,
                           const float* __restrict__ att_dst,
                           float* __restrict__ asrc, float* __restrict__ adst) {
  int t = blockIdx.x * blockDim.x + threadIdx.x;
  if (t >= N_NODES * HEADS) return;
  int head = t & 7;
  int n    = t >> 3;
  const float* __restrict__ hp = h + n * HF + head * FH;
  const float* __restrict__ as = att_src + head * FH;
  const float* __restrict__ ad = att_dst + head * FH;
  float s = 0.f, d = 0.f;
#pragma unroll
  for (int f = 0; f < FH; ++f) { float v = hp[f]; s += v * as[f]; d += v * ad[f]; }
  asrc[t] = s;
  adst[t] = d;
}

// ---------------------------------------------------------------------------
// 3) Init: out = bias (broadcast), max-keys = -inf key, denom = 0.
// ---------------------------------------------------------------------------
__global__ void gat_init(float* __restrict__ out, const float* __restrict__ bias,
                         int* __restrict__ mkey, float* __restrict__ denom) {
  int t = blockIdx.x * blockDim.x + threadIdx.x;
  if (t < N_NODES * HF) out[t] = bias[t & (HF - 1)];
  if (t < N_NODES * HEADS) { mkey[t] = (int)0x80000000; denom[t] = 0.f; }
}

// Order-preserving float<->int key (total order matches IEEE ordering).
__device__ __forceinline__ int f32_key(float f) {
  int i = __float_as_int(f);
  return i >= 0 ? i : (i ^ 0x7fffffff);
}
__device__ __forceinline__ float key_f32(int k) {
  return __int_as_float(k >= 0 ? k : (k ^ 0x7fffffff));
}

__device__ __forceinline__ float leaky(float v) {
  return v >= 0.f ? v : NEG_SLOPE * v;
}

// ---------------------------------------------------------------------------
// 4) Segment max over incoming edges (one thread per edge-head).
// ---------------------------------------------------------------------------
__global__ void gat_edge_max(const int* __restrict__ ei,
                             const float* __restrict__ asrc,
                             const float* __restrict__ adst,
                             int* __restrict__ mkey) {
  int t = blockIdx.x * blockDim.x + threadIdx.x;
  if (t >= N_EDGES * HEADS) return;
  int e = t >> 3, head = t & 7;
  int s = ei[e];
  int d = ei[N_EDGES + e];
  float v = leaky(asrc[s * HEADS + head] + adst[d * HEADS + head]);
  __hip_atomic_fetch_max(&mkey[d * HEADS + head], f32_key(v),
                         __ATOMIC_RELAXED, __HIP_MEMORY_SCOPE_AGENT);
}

// ---------------------------------------------------------------------------
// 5) ex = exp(e - m[dst]); stash per edge-head; denom[dst] += ex.
// ---------------------------------------------------------------------------
__global__ void gat_edge_exp(const int* __restrict__ ei,
                             const float* __restrict__ asrc,
                             const float* __restrict__ adst,
                             const int* __restrict__ mkey,
                             float* __restrict__ exw,
                             float* __restrict__ denom) {
  int t = blockIdx.x * blockDim.x + threadIdx.x;
  if (t >= N_EDGES * HEADS) return;
  int e = t >> 3, head = t & 7;
  int s = ei[e];
  int d = ei[N_EDGES + e];
  float v = leaky(asrc[s * HEADS + head] + adst[d * HEADS + head]);
  float m = key_f32(mkey[d * HEADS + head]);
  float ex = __expf(v - m);
  exw[t] = ex;
  __hip_atomic_fetch_add(&denom[d * HEADS + head], ex,
                         __ATOMIC_RELAXED, __HIP_MEMORY_SCOPE_AGENT);
}

// ---------------------------------------------------------------------------
// 6) out[dst] += alpha * h[src]. One wave per edge; lane covers 4 contiguous
//    features of head = lane>>2. h and out are L2-resident (25.6 MB each in
//    a 192 MB L2), so gathers + float atomics run at L2 throughput.
// ---------------------------------------------------------------------------
__global__ __launch_bounds__(256) void gat_aggregate(
    const int* __restrict__ ei, const float* __restrict__ h,
    const float* __restrict__ exw, const float* __restrict__ denom,
    float* __restrict__ out) {
  int t = blockIdx.x * blockDim.x + threadIdx.x;   // E*32 = 25.6M < 2^31
  int e    = t >> 5;
  int lane = t & 31;
  int s = ei[e];
  int d = ei[N_EDGES + e];
  int head = lane >> 2;
  float alpha = exw[e * HEADS + head] /
                (denom[d * HEADS + head] + 1e-16f);
  const v4f hv = *(const v4f*)(h + s * HF + lane * 4);
  float* __restrict__ op = out + d * HF + lane * 4;
#pragma unroll
  for (int i = 0; i < 4; ++i)
    __hip_atomic_fetch_add(&op[i], alpha * hv[i],
                           __ATOMIC_RELAXED, __HIP_MEMORY_SCOPE_AGENT);
}

// ---------------------------------------------------------------------------
extern "C" void kernel_launch(void* const* d_in, const int* in_sizes, int n_in,
                              void* d_out, int out_size, void* d_ws, size_t ws_size,
                              hipStream_t stream) {
  const float* x        = (const float*)d_in[0];   // [N, 128]
  const float* W        = (const float*)d_in[1];   // [128, 128]
  const float* att_src  = (const float*)d_in[2];   // [8, 16]
  const float* att_dst  = (const float*)d_in[3];   // [8, 16]
  const float* bias     = (const float*)d_in[4];   // [128]
  const int*   ei       = (const int*)d_in[5];     // [2, E]
  float*       out      = (float*)d_out;           // [N, 128]

  // workspace carve-up (floats)
  float* h     = (float*)d_ws;                     // N*128   = 6,400,000
  float* asrc  = h     + (size_t)N_NODES * HF;     // N*H     =   400,000
  float* adst  = asrc  + (size_t)N_NODES * HEADS;  // N*H
  float* denom = adst  + (size_t)N_NODES * HEADS;  // N*H
  int*   mkey  = (int*)(denom + (size_t)N_NODES * HEADS); // N*H ints
  float* exw   = (float*)(mkey + (size_t)N_NODES * HEADS); // E*H = 6,400,000

  const int BLK = 256;

  // 1) projection GEMM (WMMA)
  gat_gemm_wmma<<<N_NODES / 16, BLK, 0, stream>>>(x, W, h);

  // 2) attention logits
  gat_logits<<<(N_NODES * HEADS + BLK - 1) / BLK, BLK, 0, stream>>>(
      h, att_src, att_dst, asrc, adst);

  // 3) init out/keys/denom
  gat_init<<<(N_NODES * HF + BLK - 1) / BLK, BLK, 0, stream>>>(
      out, bias, mkey, denom);

  // 4) segment max
  gat_edge_max<<<(N_EDGES * HEADS + BLK - 1) / BLK, BLK, 0, stream>>>(
      ei, asrc, adst, mkey);

  // 5) exp + denom
  gat_edge_exp<<<(N_EDGES * HEADS + BLK - 1) / BLK, BLK, 0, stream>>>(
      ei, asrc, adst, mkey, exw, denom);

  // 6) weighted scatter-aggregate
  gat_aggregate<<<(N_EDGES * 32) / BLK, BLK, 0, stream>>>(
      ei, h, exw, denom, out);
}